// LearnedUpsamplingNet_8770323218643
// MI455X (gfx1250) — compile-verified
//
#include <hip/hip_runtime.h>
#include <hip/hip_bf16.h>
#include <math.h>

#define Bn 8
#define Kn 256
#define Mn 256
#define Tn 2048
#define TT 16
#define ROWP 260   // padded LDS row stride (260 % 64 banks = 4 -> conflict-free A reads)

typedef __attribute__((ext_vector_type(2))) float v2f;
typedef __attribute__((ext_vector_type(8))) float v8f;

__device__ __forceinline__ float sigmoidf_(float x) { return 1.0f / (1.0f + __expf(-x)); }

// ---------------- Stage 1: inclusive scan of d -> e, s = e - d ----------------
__global__ void scan_kernel(const float* __restrict__ d,
                            float* __restrict__ sArr, float* __restrict__ eArr) {
  int b = blockIdx.x, tid = threadIdx.x;
  __shared__ float sd[Kn];
  float dv = d[b * Kn + tid];
  sd[tid] = dv;
  __syncthreads();
  for (int off = 1; off < Kn; off <<= 1) {
    float v = (tid >= off) ? sd[tid - off] : 0.0f;
    __syncthreads();
    sd[tid] += v;
    __syncthreads();
  }
  float e = sd[tid];
  eArr[b * Kn + tid] = e;
  sArr[b * Kn + tid] = e - dv;
}

// ---------------- Stage 2: conv1d (k=3, same) + GLU for both branches ----------------
// One block per (b,k); 256 threads = input channels; cross-thread reduce for 32 outputs.
__global__ void conv_glu_kernel(const float* __restrict__ V,
                                const float* __restrict__ cw_w, const float* __restrict__ cw_b,
                                const float* __restrict__ cc_w, const float* __restrict__ cc_b,
                                float* __restrict__ Vw, float* __restrict__ Vc) {
  int b = blockIdx.x / Kn, k = blockIdx.x % Kn;
  int ci = threadIdx.x;
  const float* vr = V + ((size_t)b * Mn + ci) * Kn;
  float x0 = (k > 0) ? vr[k - 1] : 0.0f;
  float x1 = vr[k];
  float x2 = (k < Kn - 1) ? vr[k + 1] : 0.0f;
  float part[32];
#pragma unroll
  for (int co = 0; co < 16; ++co) {
    const float* w  = cw_w + ((size_t)co * Mn + ci) * 3;
    const float* wc = cc_w + ((size_t)co * Mn + ci) * 3;
    part[co]      = w[0] * x0 + w[1] * x1 + w[2] * x2;
    part[16 + co] = wc[0] * x0 + wc[1] * x1 + wc[2] * x2;
  }
#pragma unroll
  for (int c = 0; c < 32; ++c)
    for (int off = 16; off; off >>= 1)
      part[c] += __shfl_xor(part[c], off, 32);
  __shared__ float red[32][8];
  __shared__ float yf[32];
  int wave = threadIdx.x >> 5, lane = threadIdx.x & 31;
  if (lane == 0)
    for (int c = 0; c < 32; ++c) red[c][wave] = part[c];
  __syncthreads();
  if (threadIdx.x < 32) {
    int c = threadIdx.x;
    float s = 0.0f;
    for (int wv = 0; wv < 8; ++wv) s += red[c][wv];
    s += (c < 16) ? cw_b[c] : cc_b[c - 16];
    yf[c] = s;
  }
  __syncthreads();
  if (threadIdx.x < 8) {
    int c = threadIdx.x;
    Vw[((size_t)b * 8 + c) * Kn + k] = yf[c] * sigmoidf_(yf[c + 8]);
  } else if (threadIdx.x < 16) {
    int c = threadIdx.x - 8;
    Vc[((size_t)b * 8 + c) * Kn + k] = yf[16 + c] * sigmoidf_(yf[24 + c]);
  }
}

// ---------------- Stage 3: masked per-channel norm over K (in place) ----------------
__global__ void gnorm_kernel(const float* __restrict__ d,
                             const float* __restrict__ gw_w, const float* __restrict__ gw_b,
                             const float* __restrict__ gc_w, const float* __restrict__ gc_b,
                             float* __restrict__ Vw, float* __restrict__ Vc) {
  int id = blockIdx.x;
  int b = id >> 4, cs = id & 15;
  int tensor = cs >> 3, c = cs & 7;
  float* buf = tensor ? Vc : Vw;
  const float* gw = tensor ? gc_w : gw_w;
  const float* gb = tensor ? gc_b : gw_b;
  int k = threadIdx.x;
  float mk = (d[b * Kn + k] != 0.0f) ? 1.0f : 0.0f;
  float x = buf[((size_t)b * 8 + c) * Kn + k];
  __shared__ float r1[Kn], r2[Kn];
  r1[k] = x * mk;
  r2[k] = mk;
  __syncthreads();
  for (int off = 128; off; off >>= 1) {
    if (k < off) { r1[k] += r1[k + off]; r2[k] += r2[k + off]; }
    __syncthreads();
  }
  float n = r2[0];
  float mean = r1[0] / n;
  __syncthreads();
  float dx = x - mean;
  r1[k] = dx * dx * mk;
  __syncthreads();
  for (int off = 128; off; off >>= 1) {
    if (k < off) r1[k] += r1[k + off];
    __syncthreads();
  }
  float var = r1[0] / n;
  float y = dx * rsqrtf(var + 1e-5f) * gw[c] + gb[c];
  buf[((size_t)b * 8 + c) * Kn + k] = y * mk;
}

// ---------------- Stage 4: per-(b,k) layer-1 base vectors (affine-in-t trick) ----------------
__global__ void base_kernel(const float* __restrict__ sArr, const float* __restrict__ eArr,
                            const float* __restrict__ Vw, const float* __restrict__ Vc,
                            const float* __restrict__ w1, const float* __restrict__ w1b,
                            const float* __restrict__ c1, const float* __restrict__ c1b,
                            float* __restrict__ baseW, float* __restrict__ baseC) {
  int id = blockIdx.x * blockDim.x + threadIdx.x;  // 0 .. B*K-1
  int b = id / Kn, k = id % Kn;
  float s = sArr[id], e = eArr[id];
  float vw[8], vc[8];
#pragma unroll
  for (int c = 0; c < 8; ++c) {
    vw[c] = Vw[((size_t)b * 8 + c) * Kn + k];
    vc[c] = Vc[((size_t)b * 8 + c) * Kn + k];
  }
#pragma unroll
  for (int j = 0; j < 32; ++j) {
    const float* wr = w1 + j * 10;
    float acc = w1b[j] - wr[0] * s + wr[1] * e;
#pragma unroll
    for (int c = 0; c < 8; ++c) acc += wr[2 + c] * vw[c];
    baseW[(size_t)id * 32 + j] = acc;
  }
#pragma unroll
  for (int j = 0; j < 4; ++j) {
    const float* wr = c1 + j * 10;
    float acc = c1b[j] - wr[0] * s + wr[1] * e;
#pragma unroll
    for (int c = 0; c < 8; ++c) acc += wr[2 + c] * vc[c];
    baseC[(size_t)id * 4 + j] = acc;
  }
}

// ---------------- Stage 5 (fused): MLP logits -> masked softmax -> WMMA GEMM + epilogue ----------------
// Block = (b, 16-row t tile); 256 threads (8 wave32). thread -> k for the MLP phase.
__global__ void fused_kernel(const float* __restrict__ V, const float* __restrict__ d,
                             const float* __restrict__ eArr,
                             const float* __restrict__ baseW, const float* __restrict__ baseC,
                             const float* __restrict__ w1,
                             const float* __restrict__ w2, const float* __restrict__ w2b,
                             const float* __restrict__ w3, const float* __restrict__ w3b,
                             const float* __restrict__ c1,
                             const float* __restrict__ c2w, const float* __restrict__ c2b,
                             const float* __restrict__ Aw, const float* __restrict__ Ab,
                             float* __restrict__ out) {
  const int tblocks = Tn / TT;
  int b = blockIdx.x / tblocks;
  int t0 = (blockIdx.x % tblocks) * TT;
  int tid = threadIdx.x;

  __shared__ float sW[TT][ROWP];        // logits, then softmax weights (WMMA A source)
  __shared__ float sC[TT][Kn][2];       // C values per position
  __shared__ float s_w2[512], s_w2b[32], s_dW[32], s_w3[16];
  __shared__ float s_c2[8], s_c2b[4], s_dC[4];
  __shared__ float s_rowmax[TT], s_rowsum[TT];
  __shared__ float s_red[TT][16][2];
  __shared__ float s_Cs[TT][2];

  for (int i = tid; i < 512; i += 256) s_w2[i] = w2[i];
  if (tid < 32) { s_w2b[tid] = w2b[tid]; s_dW[tid] = w1[tid * 10 + 0] - w1[tid * 10 + 1]; }
  if (tid < 16) s_w3[tid] = w3[tid];
  if (tid < 8)  s_c2[tid] = c2w[tid];
  if (tid < 4)  { s_c2b[tid] = c2b[tid]; s_dC[tid] = c1[tid * 10 + 0] - c1[tid * 10 + 1]; }

  int k = tid;
  float bW[32];
#pragma unroll
  for (int j = 0; j < 32; ++j) bW[j] = baseW[((size_t)(b * Kn + k)) * 32 + j];
  float bC[4];
#pragma unroll
  for (int j = 0; j < 4; ++j) bC[j] = baseC[((size_t)(b * Kn + k)) * 4 + j];
  float mk = (d[b * Kn + k] != 0.0f) ? 1.0f : 0.0f;
  float lenT = eArr[b * Kn + Kn - 1];   // sum(d[b]) == valid T length
  float w3bias = w3b[0];
  __syncthreads();

  // --- logits + C values (layer 1 is affine in t: base + t*delta) ---
  for (int p = 0; p < TT; ++p) {
    float u = (float)(t0 + p);
    float h1[16];
#pragma unroll
    for (int j = 0; j < 16; ++j) {
      float a = bW[j] + u * s_dW[j];
      float g = bW[j + 16] + u * s_dW[j + 16];
      h1[j] = a * sigmoidf_(g);
    }
    float logit = w3bias;
#pragma unroll
    for (int jj = 0; jj < 16; ++jj) {
      float acc_a = s_w2b[jj], acc_g = s_w2b[jj + 16];
      const float* ra = s_w2 + jj * 16;
      const float* rg = s_w2 + (jj + 16) * 16;
#pragma unroll
      for (int j = 0; j < 16; ++j) { acc_a += h1[j] * ra[j]; acc_g += h1[j] * rg[j]; }
      logit += s_w3[jj] * (acc_a * sigmoidf_(acc_g));
    }
    sW[p][k] = (mk != 0.0f) ? logit : -1e9f;

    float y0 = bC[0] + u * s_dC[0], y1 = bC[1] + u * s_dC[1];
    float y2 = bC[2] + u * s_dC[2], y3 = bC[3] + u * s_dC[3];
    float hc0 = y0 * sigmoidf_(y2), hc1 = y1 * sigmoidf_(y3);
    float z[4];
#pragma unroll
    for (int jj = 0; jj < 4; ++jj) z[jj] = s_c2[jj * 2] * hc0 + s_c2[jj * 2 + 1] * hc1 + s_c2b[jj];
    sC[p][k][0] = z[0] * sigmoidf_(z[2]);
    sC[p][k][1] = z[1] * sigmoidf_(z[3]);
  }
  __syncthreads();

  // --- masked softmax over K per t-row (16 threads per row) ---
  int r = tid >> 4, j16 = tid & 15;
  float mmax = -3.0e38f;
  for (int kk = j16; kk < Kn; kk += 16) mmax = fmaxf(mmax, sW[r][kk]);
  s_red[r][j16][0] = mmax;
  __syncthreads();
  if (tid < TT) {
    float mm = s_red[tid][0][0];
    for (int i = 1; i < 16; ++i) mm = fmaxf(mm, s_red[tid][i][0]);
    s_rowmax[tid] = mm;
  }
  __syncthreads();
  float ssum = 0.0f;
  for (int kk = j16; kk < Kn; kk += 16) ssum += __expf(sW[r][kk] - s_rowmax[r]);
  s_red[r][j16][0] = ssum;
  __syncthreads();
  if (tid < TT) {
    float sm = 0.0f;
    for (int i = 0; i < 16; ++i) sm += s_red[tid][i][0];
    s_rowsum[tid] = sm;
  }
  __syncthreads();
  for (int p = 0; p < TT; ++p) {
    float wv = (mk != 0.0f) ? __expf(sW[p][k] - s_rowmax[p]) / s_rowsum[p] : 0.0f;
    sW[p][k] = wv;
  }
  __syncthreads();

  // --- Cs[t] = sum_k W[t,k] * C[t,k,:] ---
  float cs0 = 0.0f, cs1 = 0.0f;
  for (int kk = j16; kk < Kn; kk += 16) {
    float wv = sW[r][kk];
    cs0 += wv * sC[r][kk][0];
    cs1 += wv * sC[r][kk][1];
  }
  s_red[r][j16][0] = cs0;
  s_red[r][j16][1] = cs1;
  __syncthreads();
  if (tid < TT) {
    float a = 0.0f, g = 0.0f;
    for (int i = 0; i < 16; ++i) { a += s_red[tid][i][0]; g += s_red[tid][i][1]; }
    s_Cs[tid][0] = a;
    s_Cs[tid][1] = g;
  }
  __syncthreads();

  // --- O tile = W(16xK) x V^T(KxM) via chained v_wmma_f32_16x16x4_f32 ---
  // Each wave owns two N-tiles (mt = wave, wave+8) processed in ONE pass over K,
  // sharing every A-operand LDS fetch between both accumulators (2 wmma per ds load).
  int wave = tid >> 5, lane = tid & 31;
  int lrow = lane & 15, lhi = lane >> 4;
  int mcol0 = wave * 16 + lrow;            // N-tile 0
  const float* vrow0 = V + ((size_t)b * Mn + mcol0) * Kn;       // B[k][n] = V[b][n][k]
  const float* vrow1 = vrow0 + (size_t)128 * Kn;                // N-tile 1 (mcol0 + 128)
  v8f acc0 = {0.0f, 0.0f, 0.0f, 0.0f, 0.0f, 0.0f, 0.0f, 0.0f};
  v8f acc1 = {0.0f, 0.0f, 0.0f, 0.0f, 0.0f, 0.0f, 0.0f, 0.0f};
  const float* arow = &sW[lrow][0];
#pragma unroll 4
  for (int kk = 0; kk < Kn; kk += 4) {
    int ki = kk + 2 * lhi;
    v2f a2; a2.x = arow[ki]; a2.y = arow[ki + 1];      // A: M=lrow(t), K=ki..ki+1
    v2f b0; b0.x = vrow0[ki]; b0.y = vrow0[ki + 1];    // B tile 0
    v2f b1; b1.x = vrow1[ki]; b1.y = vrow1[ki + 1];    // B tile 1 (const 128KiB offset)
    acc0 = __builtin_amdgcn_wmma_f32_16x16x4_f32(false, a2, false, b0,
                                                 (short)0, acc0, false, false);
    acc1 = __builtin_amdgcn_wmma_f32_16x16x4_f32(false, a2, false, b1,
                                                 (short)0, acc1, false, false);
  }
  {
    float aw0 = Aw[mcol0 * 2], aw1 = Aw[mcol0 * 2 + 1], ab = Ab[mcol0];
    int mcol1 = mcol0 + 128;
    float bw0 = Aw[mcol1 * 2], bw1 = Aw[mcol1 * 2 + 1], bb = Ab[mcol1];
#pragma unroll
    for (int j = 0; j < 8; ++j) {            // D: M = j + 8*lhi, N = lrow
      int tl = j + 8 * lhi;
      int t = t0 + tl;
      bool valid = ((float)t < lenT);
      float o0 = acc0[j] + s_Cs[tl][0] * aw0 + s_Cs[tl][1] * aw1 + ab;
      float o1 = acc1[j] + s_Cs[tl][0] * bw0 + s_Cs[tl][1] * bw1 + bb;
      out[((size_t)b * Tn + t) * Mn + mcol0] = valid ? o0 : 0.0f;
      out[((size_t)b * Tn + t) * Mn + mcol1] = valid ? o1 : 0.0f;
    }
  }
}

extern "C" void kernel_launch(void* const* d_in, const int* in_sizes, int n_in,
                              void* d_out, int out_size, void* d_ws, size_t ws_size,
                              hipStream_t stream) {
  const float* V  = (const float*)d_in[0];
  const float* dv = (const float*)d_in[1];
  // d_in[2] = mask_K, d_in[3] = mask_T : derived from d instead (d==0 <=> invalid K; lenT = sum d)
  const float* cw_w = (const float*)d_in[4];
  const float* cw_b = (const float*)d_in[5];
  const float* gw_w = (const float*)d_in[6];
  const float* gw_b = (const float*)d_in[7];
  const float* cc_w = (const float*)d_in[8];
  const float* cc_b = (const float*)d_in[9];
  const float* gc_w = (const float*)d_in[10];
  const float* gc_b = (const float*)d_in[11];
  const float* w1  = (const float*)d_in[12];
  const float* w1b = (const float*)d_in[13];
  const float* w2  = (const float*)d_in[14];
  const float* w2b = (const float*)d_in[15];
  const float* w3  = (const float*)d_in[16];
  const float* w3b = (const float*)d_in[17];
  const float* c1  = (const float*)d_in[18];
  const float* c1b = (const float*)d_in[19];
  const float* c2w = (const float*)d_in[20];
  const float* c2b = (const float*)d_in[21];
  const float* Aw  = (const float*)d_in[22];
  const float* Ab  = (const float*)d_in[23];

  float* ws    = (float*)d_ws;
  float* sArr  = ws;                         // B*K
  float* eArr  = sArr + Bn * Kn;             // B*K
  float* Vw    = eArr + Bn * Kn;             // B*8*K
  float* Vc    = Vw + Bn * 8 * Kn;           // B*8*K
  float* baseW = Vc + Bn * 8 * Kn;           // B*K*32
  float* baseC = baseW + (size_t)Bn * Kn * 32; // B*K*4

  scan_kernel<<<dim3(Bn), dim3(Kn), 0, stream>>>(dv, sArr, eArr);
  conv_glu_kernel<<<dim3(Bn * Kn), dim3(Mn), 0, stream>>>(V, cw_w, cw_b, cc_w, cc_b, Vw, Vc);
  gnorm_kernel<<<dim3(Bn * 16), dim3(Kn), 0, stream>>>(dv, gw_w, gw_b, gc_w, gc_b, Vw, Vc);
  base_kernel<<<dim3(Bn * Kn / 256), dim3(256), 0, stream>>>(sArr, eArr, Vw, Vc, w1, w1b, c1, c1b, baseW, baseC);
  fused_kernel<<<dim3(Bn * (Tn / TT)), dim3(256), 0, stream>>>(
      V, dv, eArr, baseW, baseC, w1, w2, w2b, w3, w3b, c1, c2w, c2b, Aw, Ab, (float*)d_out);
}